// Monarch_58866821759197
// MI455X (gfx1250) — compile-verified
//
#include <hip/hip_runtime.h>

typedef __attribute__((ext_vector_type(2))) float v2f;
typedef __attribute__((ext_vector_type(8))) float v8f;

// LDS layout: one padded 64x64 fp32 matrix per token.
//   cell(t, r, c) = lds[t*ROWSTRIDE + r*INNER + c]
// INNER = 65 (inner pad) and ROWSTRIDE % 64 == 4 make every access phase
// (stage-1 store, stage-2 transposed load, stage-2 store, write-out read)
// LDS-bank-conflict free across both 16-lane halves of the wave.
static constexpr int INNER     = 65;
static constexpr int ROWSTRIDE = 64 * INNER + 4;   // 4164 floats per token
static constexpr int TB        = 16;               // tokens per workgroup

__global__ __launch_bounds__(256, 1)
void monarch_fused_kernel(const float* __restrict__ x,
                          const float* __restrict__ R,
                          const float* __restrict__ L,
                          float* __restrict__ out)
{
    __shared__ float lds[TB * ROWSTRIDE];          // ~260 KB of the 320 KB WGP LDS

    const int tid  = (int)threadIdx.x;
    const int wave = tid >> 5;                     // 8 waves (wave32)
    const int lane = tid & 31;
    const int lrow = lane & 15;                    // M / N index inside a 16-lane half
    const int lhi  = lane >> 4;                    // 0: lanes 0-15, 1: lanes 16-31
    const long tbase = (long)blockIdx.x * TB;

    //================= stage 1: Y[t][i][j] = sum_k x[t][i*64+k] * R[i][j][k] =================
    // WMMA tile: M=16 tokens, N=16 j, K=4 fp32 per instruction; 16 K-steps cover K=64.
    for (int q = 0; q < 8; ++q) {
        const int i = wave * 8 + q;                // each wave owns 8 blocks

        // A panel (16 tokens x 64 k): v2f per K-step; layout VGPR0={K0,K2}, VGPR1={K1,K3}.
        const float* ap = x + (tbase + lrow) * 4096 + i * 64 + lhi * 2;
        v2f a[16];
        #pragma unroll
        for (int kb = 0; kb < 16; ++kb)
            a[kb] = *(const v2f*)(ap + kb * 4);

        // B panel (4 k x 16 j): B[k][n] = R[i][n][k], same half-split layout.
        const float* bp = R + i * 4096 + lrow * 64 + lhi * 2;
        v8f acc0 = {}, acc1 = {}, acc2 = {}, acc3 = {};
        #pragma unroll
        for (int kb = 0; kb < 16; ++kb) {
            v2f b0 = *(const v2f*)(bp +  0 * 64 + kb * 4);
            v2f b1 = *(const v2f*)(bp + 16 * 64 + kb * 4);
            v2f b2 = *(const v2f*)(bp + 32 * 64 + kb * 4);
            v2f b3 = *(const v2f*)(bp + 48 * 64 + kb * 4);
            acc0 = __builtin_amdgcn_wmma_f32_16x16x4_f32(false, a[kb], false, b0, (short)0, acc0, false, false);
            acc1 = __builtin_amdgcn_wmma_f32_16x16x4_f32(false, a[kb], false, b1, (short)0, acc1, false, false);
            acc2 = __builtin_amdgcn_wmma_f32_16x16x4_f32(false, a[kb], false, b2, (short)0, acc2, false, false);
            acc3 = __builtin_amdgcn_wmma_f32_16x16x4_f32(false, a[kb], false, b3, (short)0, acc3, false, false);
        }

        // D layout: VGPR v holds token m = v + 8*lhi, column j = ntile*16 + lrow.
        float* yp = lds + (lhi * 8) * ROWSTRIDE + i * INNER + lrow;
        #pragma unroll
        for (int v = 0; v < 8; ++v) {
            yp[v * ROWSTRIDE +  0] = acc0[v];
            yp[v * ROWSTRIDE + 16] = acc1[v];
            yp[v * ROWSTRIDE + 32] = acc2[v];
            yp[v * ROWSTRIDE + 48] = acc3[v];
        }
    }

    __syncthreads();

    //================= stage 2: W[i][j] = sum_k Y[t][k][i] * L[i][j][k] =================
    for (int q = 0; q < 8; ++q) {
        const int i = wave * 8 + q;

        // Transposed A panel from LDS: A2[m][k] = Y[m][k*65 + i] (column i).
        const float* a2p = lds + lrow * ROWSTRIDE + (lhi * 2) * INNER + i;
        v2f a[16];
        #pragma unroll
        for (int kb = 0; kb < 16; ++kb) {
            v2f t;
            t.x = a2p[(kb * 4 + 0) * INNER];
            t.y = a2p[(kb * 4 + 1) * INNER];
            a[kb] = t;
        }

        const float* bp = L + i * 4096 + lrow * 64 + lhi * 2;
        v8f acc0 = {}, acc1 = {}, acc2 = {}, acc3 = {};
        #pragma unroll
        for (int kb = 0; kb < 16; ++kb) {
            v2f b0 = *(const v2f*)(bp +  0 * 64 + kb * 4);
            v2f b1 = *(const v2f*)(bp + 16 * 64 + kb * 4);
            v2f b2 = *(const v2f*)(bp + 32 * 64 + kb * 4);
            v2f b3 = *(const v2f*)(bp + 48 * 64 + kb * 4);
            acc0 = __builtin_amdgcn_wmma_f32_16x16x4_f32(false, a[kb], false, b0, (short)0, acc0, false, false);
            acc1 = __builtin_amdgcn_wmma_f32_16x16x4_f32(false, a[kb], false, b1, (short)0, acc1, false, false);
            acc2 = __builtin_amdgcn_wmma_f32_16x16x4_f32(false, a[kb], false, b2, (short)0, acc2, false, false);
            acc3 = __builtin_amdgcn_wmma_f32_16x16x4_f32(false, a[kb], false, b3, (short)0, acc3, false, false);
        }

        // out[t][j*64+i] = W[i][j]: store into cell [t][j*65 + i] — exactly the Y
        // column this wave just consumed for block i, so no extra barrier is needed
        // (LDS ops from one wave complete in order; other waves touch other columns).
        float* op = lds + (lhi * 8) * ROWSTRIDE + lrow * INNER + i;
        #pragma unroll
        for (int v = 0; v < 8; ++v) {
            op[v * ROWSTRIDE + 0 * 16 * INNER] = acc0[v];
            op[v * ROWSTRIDE + 1 * 16 * INNER] = acc1[v];
            op[v * ROWSTRIDE + 2 * 16 * INNER] = acc2[v];
            op[v * ROWSTRIDE + 3 * 16 * INNER] = acc3[v];
        }
    }

    __syncthreads();

    //================= coalesced write-out: 16 tokens x 4096 floats =================
    #pragma unroll 4
    for (int it = 0; it < 64; ++it) {
        const int g  = it * 1024 + tid * 4;        // flat fp32 index within the tile
        const int t  = g >> 12;                    // token within tile
        const int f  = g & 4095;                   // feature index
        const int j  = f >> 6;
        const int ic = f & 63;
        const float* p = lds + t * ROWSTRIDE + j * INNER + ic;
        float4 val = make_float4(p[0], p[1], p[2], p[3]);   // scalar LDS reads (65-pad is 4B-aligned only)
        *(float4*)(out + (tbase + t) * 4096 + f) = val;     // 16B-aligned, fully coalesced
    }
}

extern "C" void kernel_launch(void* const* d_in, const int* in_sizes, int n_in,
                              void* d_out, int out_size, void* d_ws, size_t ws_size,
                              hipStream_t stream) {
    (void)n_in; (void)out_size; (void)d_ws; (void)ws_size;
    const float* x = (const float*)d_in[0];
    const float* R = (const float*)d_in[1];
    const float* L = (const float*)d_in[2];
    float* out = (float*)d_out;

    const int tokens = in_sizes[0] / 4096;         // 16384
    const int nblocks = tokens / TB;               // 1024 workgroups
    monarch_fused_kernel<<<dim3(nblocks), dim3(256), 0, stream>>>(x, R, L, out);
}